// triggerReLU_42296837931314
// MI455X (gfx1250) — compile-verified
//
#include <hip/hip_runtime.h>
#include <math.h>
#include <stdint.h>

// CDNA5 / gfx1250: wave32, WMMA 16x16 tiles, async global->LDS loads.
typedef __attribute__((ext_vector_type(2))) float v2f;
typedef __attribute__((ext_vector_type(8))) float v8f;

#define HID   768
#define SEQ   256   // T1 == T2
#define BATCH 2
#define ROWS  (BATCH * SEQ)   // 512 rows in each GEMM
#define HPAIR (HID / 2)       // 384 rotation pairs per row

// ---------------------------------------------------------------------------
// One-shot RoPE: XR[row, 2d..2d+1] = rotate(X[row, 2d..2d+1], t * invfreq[d])
// Keeps the accurate sinf/cosf/expf slow paths out of the GEMM K-loop.
// ---------------------------------------------------------------------------
__global__ void rope_rotate(const float* __restrict__ X, float* __restrict__ XR)
{
    const int idx = blockIdx.x * blockDim.x + threadIdx.x;  // pair index
    if (idx >= ROWS * HPAIR) return;
    const int d   = idx % HPAIR;
    const int row = idx / HPAIR;
    const int t   = row & (SEQ - 1);                        // position in sequence
    const float kfreq = -2.0f * 9.210340371976184f / (float)HID; // -2*ln(1e4)/H
    const float ang = (float)t * expf((float)d * kfreq);
    const float s = sinf(ang);
    const float c = cosf(ang);
    const float2 v = ((const float2*)X)[idx];
    float2 o;
    o.x = v.x * c - v.y * s;
    o.y = v.y * c + v.x * s;
    ((float2*)XR)[idx] = o;
}

// ---------------------------------------------------------------------------
// GEMM:  C[m,n] = sum_k XR[m,k] * W[k,n]  (+ bias[n])  via V_WMMA_F32_16X16X4_F32
// Block = 256 threads = 8 waves: one 16-row m-tile, 8 adjacent 16-col n-tiles.
// A slab (16x768) staged once in LDS; +4-word row pad => conflict-free
// ds_load_2addr_b64 fragment reads. B loads are 16-lane contiguous from L2.
// ---------------------------------------------------------------------------
__global__ void gemm_wmma_f32(const float* __restrict__ XR,
                              const float* __restrict__ W,
                              const float* __restrict__ bias,
                              float* __restrict__ C)
{
    __shared__ float sA[16][HID + 4];

    const int lane  = threadIdx.x & 31;
    const int wave  = threadIdx.x >> 5;
    const int mtile = blockIdx.x;                 // 0..31   (512/16)
    const int ntile = blockIdx.y * 8 + wave;      // 0..47   (768/16)

#pragma unroll
    for (int e = 0; e < (16 * HID) / (256 * 4); ++e) {
        const int lin = (threadIdx.x + e * 256) * 4;
        const int r = lin / HID, cc = lin % HID;
        *(float4*)&sA[r][cc] =
            *(const float4*)&XR[(size_t)(mtile * 16 + r) * HID + cc];
    }
    __syncthreads();

    const int arow = lane & 15;
    const int kk   = 2 * (lane >> 4);             // lane's k offset in K-slab
    const int n    = ntile * 16 + (lane & 15);    // B/C column for this lane
    const float* __restrict__ Wn = W + n;

    v8f acc = {};
#pragma unroll 4
    for (int k0 = 0; k0 < HID; k0 += 4) {
        const int k = k0 + kk;
        const v2f a = *(const v2f*)&sA[arow][k];
        v2f b;
        b.x = Wn[(size_t)(k    ) * HID];
        b.y = Wn[(size_t)(k + 1) * HID];
        acc = __builtin_amdgcn_wmma_f32_16x16x4_f32(false, a, false, b,
                                                    (short)0, acc, false, false);
    }

    const float bn = bias ? bias[n] : 0.0f;
    const int rbase = mtile * 16 + 8 * (lane >> 4);
#pragma unroll
    for (int v = 0; v < 8; ++v) {
        C[(size_t)(rbase + v) * HID + n] = acc[v] + bn;
    }
}

// ---------------------------------------------------------------------------
// Pairwise ReLU-reduce + sigmoid (dominant phase, inherently VALU):
//   out[b,i,j] = sigmoid( b_tri + sum_h relu(ax[b,i,h] + by[b,j,h]) * w[h] )
// Double-buffered staging with CDNA5 GLOBAL_LOAD_ASYNC_TO_LDS_B128:
// each wave issues 4 async b128 per chunk (its eighth of the 16x128 ax and
// by slabs); async loads complete in order, so s_wait_asynccnt 4 after issuing
// the next chunk guarantees the current chunk has landed while the next is
// still in flight. +4-word row pad => the 16 distinct-row float4 reads cover
// all 64 LDS banks; same-row reads broadcast.
// ---------------------------------------------------------------------------
#define CH      128
#define NCHUNK  (HID / CH)   // 6

__device__ __forceinline__ void issue_async_chunk(
    const float* __restrict__ AX, const float* __restrict__ BY,
    int rowA0, int rowB0, int h0,
    float (*bax)[CH + 4], float (*bby)[CH + 4], int tid)
{
#pragma unroll
    for (int e = 0; e < 2; ++e) {
        const int lin = tid + e * 256;            // float4 slot 0..511
        const int r = lin >> 5;                   // 32 float4 per row
        const int c = (lin & 31) * 4;
        const unsigned la = (unsigned)(uintptr_t)&bax[r][c];
        const float*   ga = &AX[(size_t)(rowA0 + r) * HID + h0 + c];
        asm volatile("global_load_async_to_lds_b128 %0, %1, off"
                     :: "v"(la), "v"(ga) : "memory");
        const unsigned lb = (unsigned)(uintptr_t)&bby[r][c];
        const float*   gb = &BY[(size_t)(rowB0 + r) * HID + h0 + c];
        asm volatile("global_load_async_to_lds_b128 %0, %1, off"
                     :: "v"(lb), "v"(gb) : "memory");
    }
}

__global__ void pair_relu_sigmoid(const float* __restrict__ AX,
                                  const float* __restrict__ BY,
                                  const float* __restrict__ wtri,
                                  const float* __restrict__ btri,
                                  float* __restrict__ out)
{
    __shared__ float sax[2][16][CH + 4];
    __shared__ float sby[2][16][CH + 4];
    __shared__ float sw[HID];

    const int tid = threadIdx.x;                  // 0..255
    const int blk = blockIdx.x;                   // 0..511
    const int bi  = blk >> 8;                     // batch
    const int i0  = ((blk >> 4) & 15) * 16;
    const int j0  = (blk & 15) * 16;
    const int il  = tid >> 4;                     // local i (2 rows/wave: bcast)
    const int jl  = tid & 15;                     // local j (16 rows: padded)
    const int rowA0 = bi * SEQ + i0;
    const int rowB0 = bi * SEQ + j0;

    // Stage w_tri once (3 KB); covered by the first barrier below.
    if (tid < HID / 4)
        *(float4*)&sw[tid * 4] = *(const float4*)&wtri[tid * 4];

    // Prologue: chunk 0 -> buffer 0.
    issue_async_chunk(AX, BY, rowA0, rowB0, 0, sax[0], sby[0], tid);

    float acc = 0.0f;
    int bsel = 0;
#pragma unroll
    for (int cI = 0; cI < NCHUNK; ++cI) {
        if (cI + 1 < NCHUNK) {
            issue_async_chunk(AX, BY, rowA0, rowB0, (cI + 1) * CH,
                              sax[bsel ^ 1], sby[bsel ^ 1], tid);
            // 8 outstanding, in-order completion: <=4 left => chunk cI landed.
            asm volatile("s_wait_asynccnt 0x4" ::: "memory");
        } else {
            asm volatile("s_wait_asynccnt 0x0" ::: "memory");
        }
        __syncthreads();                          // chunk cI visible to all waves

        const int h0 = cI * CH;
#pragma unroll 8
        for (int h = 0; h < CH; h += 4) {
            const float4 a4 = *(const float4*)&sax[bsel][il][h];
            const float4 b4 = *(const float4*)&sby[bsel][jl][h];
            const float4 w4 = *(const float4*)&sw[h0 + h];
            acc = fmaf(fmaxf(a4.x + b4.x, 0.0f), w4.x, acc);
            acc = fmaf(fmaxf(a4.y + b4.y, 0.0f), w4.y, acc);
            acc = fmaf(fmaxf(a4.z + b4.z, 0.0f), w4.z, acc);
            acc = fmaf(fmaxf(a4.w + b4.w, 0.0f), w4.w, acc);
        }
        __syncthreads();  // all reads of buf[bsel] done before chunk cI+2 overwrites
        bsel ^= 1;
    }

    const float o = acc + btri[0];
    out[((size_t)(bi * SEQ + i0 + il) * SEQ) + j0 + jl] = 1.0f / (1.0f + expf(-o));
}

// ---------------------------------------------------------------------------
// Inputs (setup_inputs order): x, y, W1, W2, b, w_tri, b_tri  (all float32)
// Output: [B, T1*T2, 1] float32 = 131072 elements.
// Workspace (4.5 MiB): [buf0: xr -> reused as by][buf1: yr][buf2: ax]
// ---------------------------------------------------------------------------
extern "C" void kernel_launch(void* const* d_in, const int* in_sizes, int n_in,
                              void* d_out, int out_size, void* d_ws, size_t ws_size,
                              hipStream_t stream)
{
    const float* x  = (const float*)d_in[0];
    const float* y  = (const float*)d_in[1];
    const float* W1 = (const float*)d_in[2];
    const float* W2 = (const float*)d_in[3];
    const float* b  = (const float*)d_in[4];
    const float* wt = (const float*)d_in[5];
    const float* bt = (const float*)d_in[6];
    float* out = (float*)d_out;

    const size_t NBUF = (size_t)ROWS * HID;       // 512*768 floats
    float* xr = (float*)d_ws;                     // buf0
    float* yr = xr + NBUF;                        // buf1
    float* ax = yr + NBUF;                        // buf2
    float* by = xr;                               // alias buf0 (xr dead by then)

    const int npair = ROWS * HPAIR;               // 196608 pairs per tensor
    rope_rotate<<<(npair + 255) / 256, 256, 0, stream>>>(x, xr);
    rope_rotate<<<(npair + 255) / 256, 256, 0, stream>>>(y, yr);

    const dim3 gemm_grid(ROWS / 16, HID / 128);   // (32, 6), 8 n-tiles/block
    gemm_wmma_f32<<<gemm_grid, 256, 0, stream>>>(xr, W1, nullptr, ax);
    gemm_wmma_f32<<<gemm_grid, 256, 0, stream>>>(yr, W2, b, by);

    pair_relu_sigmoid<<<BATCH * (SEQ / 16) * (SEQ / 16), 256, 0, stream>>>(
        ax, by, wt, bt, out);
}